// OrthogonalLayer_21174188769397
// MI455X (gfx1250) — compile-verified
//
#include <hip/hip_runtime.h>

typedef float v2f __attribute__((ext_vector_type(2)));
typedef float v8f __attribute__((ext_vector_type(8)));

#define NUM_IT 10
#define NS_IT  8
#define M_DIM  64
#define N_DIM  32
#define LDQ    33   // padded LDS stride (64x32 matrices)
#define LDG    33   // padded LDS stride (32x32 matrices)

__device__ __forceinline__ v8f wmma4(v2f a, v2f b, v8f c) {
  // D = A(16x4,f32) * B(4x16,f32) + C(16x16,f32)
  return __builtin_amdgcn_wmma_f32_16x16x4_f32(false, a, false, b, (short)0, c, false, false);
}

// A fragment (16x4) from row-major X: rows row0..row0+15, cols k0..k0+3
// lane layout: VGPR0 = K (k0 + 2*(lane>>4)), VGPR1 = K+1
__device__ __forceinline__ v2f fragA(const float* X, int ld, int row0, int k0, int lane) {
  int r = row0 + (lane & 15);
  int k = k0 + ((lane >> 4) << 1);
  v2f a;
  a.x = X[r * ld + k];
  a.y = X[r * ld + k + 1];
  return a;
}

// A^T fragment: logical A[m][k] = X[k][row0+m]  (for Q^H Q)
__device__ __forceinline__ v2f fragAT(const float* X, int ld, int row0, int k0, int lane) {
  int r = row0 + (lane & 15);
  int k = k0 + ((lane >> 4) << 1);
  v2f a;
  a.x = X[k * ld + r];
  a.y = X[(k + 1) * ld + r];
  return a;
}

// B fragment (4x16): rows k0..k0+3, cols col0..col0+15
__device__ __forceinline__ v2f fragB(const float* X, int ld, int k0, int col0, int lane) {
  int c = col0 + (lane & 15);
  int k = k0 + ((lane >> 4) << 1);
  v2f b;
  b.x = X[k * ld + c];
  b.y = X[(k + 1) * ld + c];
  return b;
}

// C/D 16x16 tile store: VGPR v -> row (m0+v), lanes>=16 -> rows +8
__device__ __forceinline__ void storeC(float* X, int ld, int row0, int col0, int lane,
                                       v8f c, float scale) {
  int n  = col0 + (lane & 15);
  int m0 = row0 + ((lane >> 4) << 3);
#pragma unroll
  for (int v = 0; v < 8; ++v) X[(m0 + v) * ld + n] = c[v] * scale;
}

__global__ __launch_bounds__(128)
void ortho_ns_kernel(const float* __restrict__ x, float* __restrict__ out) {
  __shared__ float sQr[M_DIM * LDQ];
  __shared__ float sQi[M_DIM * LDQ];
  __shared__ float sGr[N_DIM * LDG];
  __shared__ float sGi[N_DIM * LDG];
  __shared__ float sYr[N_DIM * LDG];
  __shared__ float sYi[N_DIM * LDG];
  __shared__ float sZr[N_DIM * LDG];
  __shared__ float sZi[N_DIM * LDG];
  __shared__ float sTr[N_DIM * LDG];
  __shared__ float sTi[N_DIM * LDG];
  __shared__ float sScale[2];

  const int tid  = threadIdx.x;
  const int lane = tid & 31;
  const int wave = tid >> 5;
  const size_t b = blockIdx.x;

  // ---- load Q (re/im interleaved) into LDS ----
  const float2* xin = (const float2*)(x + b * (size_t)(M_DIM * N_DIM * 2));
#pragma unroll
  for (int e = tid; e < M_DIM * N_DIM; e += 128) {
    float2 v = xin[e];
    int m = e >> 5, n = e & 31;
    sQr[m * LDQ + n] = v.x;
    sQi[m * LDQ + n] = v.y;
  }
  __syncthreads();

  const int ti = (wave >> 1) << 4;  // 0/16: row tile of 32x32 outputs
  const int tj = (wave & 1) << 4;   // 0/16: col tile

  for (int it = 0; it < NUM_IT; ++it) {
    // ================= G = Q^H Q (32x32 complex, Hermitian) =================
    // Gr = Qr^T Qr + Qi^T Qi ;  Gi = Qr^T Qi - Qi^T Qr
    v8f gR = {}; v8f gI = {};
#pragma unroll
    for (int k0 = 0; k0 < M_DIM; k0 += 4) {
      v2f aR = fragAT(sQr, LDQ, ti, k0, lane);
      v2f aI = fragAT(sQi, LDQ, ti, k0, lane);
      v2f bR = fragB(sQr, LDQ, k0, tj, lane);
      v2f bI = fragB(sQi, LDQ, k0, tj, lane);
      gR = wmma4(aR, bR, gR);
      gR = wmma4(aI, bI, gR);
      gI = wmma4(aR, bI, gI);
      v2f nI = -aI;
      gI = wmma4(nI, bR, gI);
    }
    storeC(sGr, LDG, ti, tj, lane, gR, 1.0f);
    storeC(sGi, LDG, ti, tj, lane, gI, 1.0f);
    __syncthreads();

    // ---- normalization scale: trace(G) = ||Q||_F^2 (real, >0) ----
    if (tid == 0) {
      float tr = 0.0f;
      for (int i = 0; i < N_DIM; ++i) tr += sGr[i * LDG + i];
      tr = fmaxf(tr, 1e-8f);
      sScale[0] = 1.0f / tr;
      sScale[1] = rsqrtf(tr);
    }
    __syncthreads();

    // ---- Y = G / trace(G), Z = I ----
    float invs = sScale[0];
#pragma unroll
    for (int e = tid; e < N_DIM * N_DIM; e += 128) {
      int i = e >> 5, j = e & 31;
      sYr[i * LDG + j] = sGr[i * LDG + j] * invs;
      sYi[i * LDG + j] = sGi[i * LDG + j] * invs;
      sZr[i * LDG + j] = (i == j) ? 1.0f : 0.0f;
      sZi[i * LDG + j] = 0.0f;
    }
    __syncthreads();

    // ================= Newton-Schulz: Z -> (G/tr)^(-1/2) =================
    for (int ns = 0; ns < NS_IT; ++ns) {
      // T = 1.5*I - 0.5 * Z*Y  (complex 32x32, K=32)
      v8f pR = {}; v8f pI = {};
#pragma unroll
      for (int k0 = 0; k0 < N_DIM; k0 += 4) {
        v2f zr = fragA(sZr, LDG, ti, k0, lane);
        v2f zi = fragA(sZi, LDG, ti, k0, lane);
        v2f yr = fragB(sYr, LDG, k0, tj, lane);
        v2f yi = fragB(sYi, LDG, k0, tj, lane);
        pR = wmma4(zr, yr, pR);
        v2f nzi = -zi;
        pR = wmma4(nzi, yi, pR);
        pI = wmma4(zr, yi, pI);
        pI = wmma4(zi, yr, pI);
      }
      {
        int n  = tj + (lane & 15);
        int m0 = ti + ((lane >> 4) << 3);
#pragma unroll
        for (int v = 0; v < 8; ++v) {
          int m = m0 + v;
          sTr[m * LDG + n] = ((m == n) ? 1.5f : 0.0f) - 0.5f * pR[v];
          sTi[m * LDG + n] = -0.5f * pI[v];
        }
      }
      __syncthreads();

      // Ynew = Y*T ; Znew = T*Z  (accumulate in regs, barrier, then overwrite)
      v8f yR = {}; v8f yI = {}; v8f zR = {}; v8f zI = {};
#pragma unroll
      for (int k0 = 0; k0 < N_DIM; k0 += 4) {
        v2f ar = fragA(sYr, LDG, ti, k0, lane);
        v2f ai = fragA(sYi, LDG, ti, k0, lane);
        v2f br = fragB(sTr, LDG, k0, tj, lane);
        v2f bi = fragB(sTi, LDG, k0, tj, lane);
        yR = wmma4(ar, br, yR);
        v2f nai = -ai;
        yR = wmma4(nai, bi, yR);
        yI = wmma4(ar, bi, yI);
        yI = wmma4(ai, br, yI);

        v2f cr = fragA(sTr, LDG, ti, k0, lane);
        v2f ci = fragA(sTi, LDG, ti, k0, lane);
        v2f dr = fragB(sZr, LDG, k0, tj, lane);
        v2f di = fragB(sZi, LDG, k0, tj, lane);
        zR = wmma4(cr, dr, zR);
        v2f nci = -ci;
        zR = wmma4(nci, di, zR);
        zI = wmma4(cr, di, zI);
        zI = wmma4(ci, dr, zI);
      }
      __syncthreads();  // all waves done reading Y,Z
      storeC(sYr, LDG, ti, tj, lane, yR, 1.0f);
      storeC(sYi, LDG, ti, tj, lane, yI, 1.0f);
      storeC(sZr, LDG, ti, tj, lane, zR, 1.0f);
      storeC(sZi, LDG, ti, tj, lane, zI, 1.0f);
      __syncthreads();
    }

    // ================= Q = Q * Z * rsqrt(trace) =================
    // wave w owns rows [16w, 16w+16): two 16x16 column tiles, K=32
    float rs = sScale[1];
    const int r0 = wave << 4;
    v8f q0R = {}; v8f q0I = {}; v8f q1R = {}; v8f q1I = {};
#pragma unroll
    for (int k0 = 0; k0 < N_DIM; k0 += 4) {
      v2f ar  = fragA(sQr, LDQ, r0, k0, lane);
      v2f ai  = fragA(sQi, LDQ, r0, k0, lane);
      v2f nai = -ai;
      v2f b0r = fragB(sZr, LDG, k0, 0, lane);
      v2f b0i = fragB(sZi, LDG, k0, 0, lane);
      q0R = wmma4(ar, b0r, q0R);
      q0R = wmma4(nai, b0i, q0R);
      q0I = wmma4(ar, b0i, q0I);
      q0I = wmma4(ai, b0r, q0I);
      v2f b1r = fragB(sZr, LDG, k0, 16, lane);
      v2f b1i = fragB(sZi, LDG, k0, 16, lane);
      q1R = wmma4(ar, b1r, q1R);
      q1R = wmma4(nai, b1i, q1R);
      q1I = wmma4(ar, b1i, q1I);
      q1I = wmma4(ai, b1r, q1I);
    }
    // rows are exclusive per wave; loads above precede stores in-order
    storeC(sQr, LDQ, r0, 0,  lane, q0R, rs);
    storeC(sQi, LDQ, r0, 0,  lane, q0I, rs);
    storeC(sQr, LDQ, r0, 16, lane, q1R, rs);
    storeC(sQi, LDQ, r0, 16, lane, q1I, rs);
    __syncthreads();
  }

  // ---- write output (re/im interleaved) ----
  float2* o = (float2*)(out + b * (size_t)(M_DIM * N_DIM * 2));
#pragma unroll
  for (int e = tid; e < M_DIM * N_DIM; e += 128) {
    int m = e >> 5, n = e & 31;
    float2 v;
    v.x = sQr[m * LDQ + n];
    v.y = sQi[m * LDQ + n];
    o[e] = v;
  }
}

extern "C" void kernel_launch(void* const* d_in, const int* in_sizes, int n_in,
                              void* d_out, int out_size, void* d_ws, size_t ws_size,
                              hipStream_t stream) {
  (void)n_in; (void)d_ws; (void)ws_size; (void)out_size;
  const float* x = (const float*)d_in[0];
  float* out = (float*)d_out;
  int batches = in_sizes[0] / (M_DIM * N_DIM * 2);  // 8192
  ortho_ns_kernel<<<batches, 128, 0, stream>>>(x, out);
}